// BGRUClassifier_88399016886506
// MI455X (gfx1250) — compile-verified
//
#include <hip/hip_runtime.h>
#include <hip/hip_bf16.h>
#include <math.h>

// ---------------------------------------------------------------------------
// BGRU classifier for MI455X (gfx1250, wave32).
//   T=84, B=96, E=400, H=10, L=10 bidirectional layers, head -> (96,19) f32.
//
//   * parallel matmul work (per-layer input projections, M=8064) goes through
//     V_WMMA_F32_16X16X4_F32 (f32 WMMA = bit-parity with the f32 reference)
//   * the serial 10x84-step scan uses 2 blocks x 960 threads: thread (b,j)
//     owns one hidden unit; h is double-buffered in LDS so each step costs
//     ~70 VALU ops + ONE workgroup barrier instead of ~700 serial VALU ops.
// ---------------------------------------------------------------------------

typedef __attribute__((ext_vector_type(2))) float v2f;
typedef __attribute__((ext_vector_type(8))) float v8f;

#define T_STEPS 84
#define BATCH   96
#define HID     10
#define GATES   30            // 3*HID per direction
#define NCOLS   60            // 2 directions * 30 gates
#define XPT_STRIDE 64         // padded column stride of xpT
#define M_TOTAL (T_STEPS*BATCH)   // 8064 rows, = 504 tiles of 16

// ---------------------------------------------------------------------------
// GEMM: xpT[t][col][b] = bias[col] + sum_k A[row(t,b)][k] * W[col][k]
//   A: emb (gather via idx) for layer 0 (K=400), or prev layer out (K=20).
//   W: (60 x K) = w_ih for both directions stacked.
// One block = one 16-row M tile; 4 waves each own a 16-col N tile.
// WMMA f32 16x16x4 layouts (wave32):
//   A lane L: row = L%16; VGPR0/1 hold K = {0,1} (L<16) or {2,3} (L>=16)
//   B lane L: col = L%16; VGPR0/1 hold K = {0,1} (L<16) or {2,3} (L>=16)
//   C/D VGPR v: row = v + 8*(L>=16), col = L%16
// ---------------------------------------------------------------------------
__global__ __launch_bounds__(128)
void gemm_xp_kernel(const float* __restrict__ A,
                    const int*   __restrict__ idx,   // nullptr for layers > 0
                    int K,
                    const float* __restrict__ W,     // (60 x K)
                    const float* __restrict__ bias,  // (60)
                    float* __restrict__ xpT)         // [84][64][96]
{
    extern __shared__ float Atile[];                 // 16 x (K+2), padded vs bank conflicts
    const int strideA = K + 2;
    const int mtile   = blockIdx.x;                  // 0..503
    const int tid     = threadIdx.x;

    // Cooperative A-tile load (embedding gather for layer 0).
    for (int e = tid; e < 16 * K; e += 128) {
        int m = e / K, k = e - m * K;
        int row  = mtile * 16 + m;                   // < 8064 always (504*16)
        int srow = idx ? idx[row] : row;
        Atile[m * strideA + k] = A[(long)srow * K + k];
    }
    __syncthreads();

    const int wave = tid >> 5;
    const int lane = tid & 31;
    const int half = lane >> 4;                      // 0: lanes 0-15, 1: 16-31
    const int l16  = lane & 15;
    const int off  = half * 2;                       // K sub-offset in the 4-step
    const int col  = wave * 16 + l16;                // 0..63
    const int wcol = (col < NCOLS) ? col : 0;        // clamp: keep loads in-bounds,
                                                     // keep EXEC uniform for WMMA
    const float* Wrow = W + (long)wcol * K;
    const float* Arow = Atile + l16 * strideA;

    v8f c = {};
    for (int kb = 0; kb < K; kb += 4) {
        v2f a = *(const v2f*)(Arow + kb + off);      // A[m][kb+off], [kb+off+1]
        v2f b = *(const v2f*)(Wrow + kb + off);      // W[col][kb+off], [kb+off+1]
        c = __builtin_amdgcn_wmma_f32_16x16x4_f32(false, a, false, b,
                                                  (short)0, c, false, false);
    }

    if (col < NCOLS) {
        float bv = bias[col];
        #pragma unroll
        for (int v = 0; v < 8; ++v) {
            int m = mtile * 16 + v + half * 8;       // global row
            int t = m / BATCH, b = m - t * BATCH;
            xpT[((long)t * XPT_STRIDE + col) * BATCH + b] = c[v] + bv;
        }
    }
}

// ---------------------------------------------------------------------------
// Sequential GRU scan for one layer, both directions in parallel.
// grid = 2 (direction), block = 960: tid = j*96 + b, thread owns hidden unit
// (b, j).  Per step: 3 dot products of length 10 + gate math, then publish
// h_new through a double-buffered LDS array (one barrier per step).
//   r = sig(xr+hr); z = sig(xz+hz); n = tanh(xn + r*hn); h = (1-z)*n + z*h
// ---------------------------------------------------------------------------
__global__ __launch_bounds__(960)
void gru_scan_kernel(const float* __restrict__ xpT,   // [84][64][96]
                     const float* __restrict__ w_hh,  // (2,30,10)
                     const float* __restrict__ b_hh,  // (2,30)
                     float* __restrict__ out)         // [8064][20]
{
    __shared__ float whh[GATES * HID];
    __shared__ float bhh[GATES];
    __shared__ float hbuf[2][HID * BATCH];           // [i*96 + b], double-buffered
    const int dir = blockIdx.x;
    const int tid = threadIdx.x;                     // = j*96 + b
    const int j   = tid / BATCH;                     // hidden unit 0..9
    const int b   = tid - j * BATCH;                 // batch element 0..95

    for (int e = tid; e < GATES * HID; e += HID * BATCH)
        whh[e] = w_hh[dir * GATES * HID + e];
    if (tid < GATES) bhh[tid] = b_hh[dir * GATES + tid];
    hbuf[0][tid] = 0.f;                              // h0 = 0 (read at step 0)
    __syncthreads();

    // Per-thread constant weight rows (registers): gates r=j, z=10+j, n=20+j
    float wr[HID], wz[HID], wn[HID];
    #pragma unroll
    for (int i = 0; i < HID; ++i) {
        wr[i] = whh[(j)            * HID + i];
        wz[i] = whh[(HID + j)      * HID + i];
        wn[i] = whh[(2 * HID + j)  * HID + i];
    }
    const float br = bhh[j], bz = bhh[HID + j], bn = bhh[2 * HID + j];

    for (int s = 0; s < T_STEPS; ++s) {
        const int t  = dir ? (T_STEPS - 1 - s) : s;  // backward scans t high->low
        const int rd = s & 1;                        // read buffer
        const float* hr = &hbuf[rd][b];

        float gr = br, gz = bz, gn = bn;
        #pragma unroll
        for (int i = 0; i < HID; ++i) {
            float hi = hr[i * BATCH];
            gr += hi * wr[i];
            gz += hi * wz[i];
            gn += hi * wn[i];
        }
        float hold = hr[j * BATCH];

        const float* xp = xpT + ((long)t * XPT_STRIDE + dir * GATES) * BATCH + b;
        float xr = xp[(long)(j)           * BATCH];
        float xz = xp[(long)(HID + j)     * BATCH];
        float xn = xp[(long)(2 * HID + j) * BATCH];

        float r    = 1.f / (1.f + expf(-(xr + gr)));
        float z    = 1.f / (1.f + expf(-(xz + gz)));
        float n    = tanhf(xn + r * gn);
        float hnew = (1.f - z) * n + z * hold;

        hbuf[rd ^ 1][tid] = hnew;                    // publish to other buffer
        out[((long)t * BATCH + b) * (2 * HID) + dir * HID + j] = hnew;
        __syncthreads();                             // one barrier per step
    }
}

// ---------------------------------------------------------------------------
// Pooling: out[:, :96, :18] -> (84,32,3,6,3) -> mean/max over (2,4) ->
// cat -> reshape to feat[96][336]. One thread per (t,bg,f) pair (16128).
// ---------------------------------------------------------------------------
__global__ void pool_kernel(const float* __restrict__ out,   // [8064][20]
                            float* __restrict__ feat)        // [96][336]
{
    int i = blockIdx.x * blockDim.x + threadIdx.x;
    if (i >= T_STEPS * 32 * 6) return;
    int t   = i / 192;
    int rem = i - t * 192;
    int bg  = rem / 6;
    int f   = rem - bg * 6;
    float s = 0.f, mx = -1e30f;
    for (int ib = 0; ib < 3; ++ib)
        for (int ic = 0; ic < 3; ++ic) {
            float v = out[((long)t * BATCH + 3 * bg + ib) * 20 + 3 * f + ic];
            s += v;
            mx = fmaxf(mx, v);
        }
    int q  = t * 32 + bg;                 // 12-block index, 28 blocks per row
    int r  = q / 28;
    int cb = (q - r * 28) * 12;
    feat[r * 336 + cb + f]     = s * (1.f / 9.f);
    feat[r * 336 + cb + 6 + f] = mx;
}

// ---------------------------------------------------------------------------
// Head: sigmoid(feat @ lin_w^T + lin_b) -> (96,19)
// ---------------------------------------------------------------------------
__global__ void head_kernel(const float* __restrict__ feat,  // [96][336]
                            const float* __restrict__ lw,    // (19,336)
                            const float* __restrict__ lb,    // (19)
                            float* __restrict__ outp)        // (96,19)
{
    int i = blockIdx.x * blockDim.x + threadIdx.x;
    if (i >= 96 * 19) return;
    int r = i / 19, c = i - r * 19;
    float s = lb[c];
    const float* fr = feat + (long)r * 336;
    const float* wr = lw + (long)c * 336;
    for (int k = 0; k < 336; ++k) s += fr[k] * wr[k];
    outp[i] = 1.f / (1.f + expf(-s));
}

// ---------------------------------------------------------------------------
extern "C" void kernel_launch(void* const* d_in, const int* in_sizes, int n_in,
                              void* d_out, int out_size, void* d_ws, size_t ws_size,
                              hipStream_t stream)
{
    const int*   x     = (const int*)  d_in[0];   // (84,96) int32 token ids
    const float* emb   = (const float*)d_in[1];   // (100000,400)
    const float* w_ih0 = (const float*)d_in[2];   // (2,30,400)
    const float* w_hh0 = (const float*)d_in[3];   // (2,30,10)
    const float* b_ih0 = (const float*)d_in[4];   // (2,30)
    const float* b_hh0 = (const float*)d_in[5];   // (2,30)
    const float* w_ih  = (const float*)d_in[6];   // (9,2,30,20)
    const float* w_hh  = (const float*)d_in[7];   // (9,2,30,10)
    const float* b_ih  = (const float*)d_in[8];   // (9,2,30)
    const float* b_hh  = (const float*)d_in[9];   // (9,2,30)
    const float* lin_w = (const float*)d_in[10];  // (19,336)
    const float* lin_b = (const float*)d_in[11];  // (19)

    // Workspace carve-up (f32): xpT | outA | outB | feat  (~3.4 MB total)
    float* xpT  = (float*)d_ws;                          // 84*64*96
    float* outA = xpT  + (size_t)T_STEPS * XPT_STRIDE * BATCH;
    float* outB = outA + (size_t)M_TOTAL * 20;
    float* feat = outB + (size_t)M_TOTAL * 20;           // 96*336

    // Layer 0: fused embedding gather + input projection (K=400) via WMMA.
    gemm_xp_kernel<<<M_TOTAL / 16, 128, (400 + 2) * 16 * sizeof(float), stream>>>(
        emb, x, 400, w_ih0, b_ih0, xpT);
    gru_scan_kernel<<<2, HID * BATCH, 0, stream>>>(xpT, w_hh0, b_hh0, outA);

    // Layers 1..9: input projection (K=20) via WMMA, then scan; ping-pong.
    float* cur = outA;
    float* nxt = outB;
    for (int l = 1; l < 10; ++l) {
        gemm_xp_kernel<<<M_TOTAL / 16, 128, (20 + 2) * 16 * sizeof(float), stream>>>(
            cur, nullptr, 20,
            w_ih + (size_t)(l - 1) * 60 * 20,
            b_ih + (size_t)(l - 1) * 60, xpT);
        gru_scan_kernel<<<2, HID * BATCH, 0, stream>>>(
            xpT,
            w_hh + (size_t)(l - 1) * 60 * 10,
            b_hh + (size_t)(l - 1) * 60, nxt);
        float* tmp = cur; cur = nxt; nxt = tmp;
    }
    // cur == layer-9 output

    pool_kernel<<<(T_STEPS * 32 * 6 + 127) / 128, 128, 0, stream>>>(cur, feat);
    head_kernel<<<(96 * 19 + 127) / 128, 128, 0, stream>>>(feat, lin_w, lin_b,
                                                           (float*)d_out);
}